// MCPBRNN_Generic_constant_Out_variableLoss_10196252360837
// MI455X (gfx1250) — compile-verified
//
#include <hip/hip_runtime.h>
#include <math.h>

#define BB 5000
#define TT 10
#define NTHREADS 1024
#define NCHUNK ((BB + NTHREADS - 1) / NTHREADS)   // 5

static __device__ constexpr float ML = 2.9086f;
static __device__ constexpr float SL = 1.898f;

typedef __attribute__((ext_vector_type(2))) float v2f;
typedef __attribute__((ext_vector_type(8))) float v8f;

// Single persistent-workgroup kernel: chunked affine scan over b, fully fused.
//   c <- f*c + g flattened over B*T steps; "active" depends only on b, so each
//   b contributes one affine map (F,G). We scan those maps (exclusive) to get
//   the incoming state per b, then apply the 9-step partial (F9,G9).
__global__ void __launch_bounds__(NTHREADS)
fused_mcpbrnn(const float* __restrict__ x,
              const float* __restrict__ y_obs,
              const int*   __restrict__ time_lag,
              const float* __restrict__ w_yom,
              const float* __restrict__ w_ylm,
              const float* __restrict__ w_yfm,
              const float* __restrict__ b0_ylm,
              const float* __restrict__ w_b2,
              const int*   __restrict__ spinLen,
              const int*   __restrict__ trainLen,
              float*       __restrict__ out)
{
    __shared__ float ybuf[4096];               // y_obs[0:4096) staged via async DMA
    __shared__ float wAF[32], wAG[32];         // wave aggregates
    __shared__ float wPF[32], wPG[32];         // wave exclusive prefixes (w/ carry)
    __shared__ float wsum[32], wsq[32];        // std partials
    __shared__ float sc[3];                    // carryF, carryG, obsstd

    const int tid  = threadIdx.x;
    const int lane = tid & 31;
    const int wave = tid >> 5;

    // ---- stage y_obs[0:4096) into LDS with async global->LDS DMA ----
    {
        unsigned dst = (unsigned)(unsigned long long)(void*)ybuf + 16u * (unsigned)tid;
        unsigned src = 16u * (unsigned)tid;    // 1024 lanes x 16B = 16 KB
        asm volatile("global_load_async_to_lds_b128 %0, %1, %2 offset:0"
                     :: "v"(dst), "v"(src), "s"(y_obs) : "memory");
    }

    // ---- scalar parameters (computed while DMA is in flight) ----
    const float eo = expf(w_yom[0]), el = expf(w_ylm[0]), ef = expf(w_yfm[0]);
    const float denom = eo + el + ef;
    const float oo = eo / denom, ol1 = el / denom;
    const float bias = b0_ylm[0], wb2 = w_b2[0];
    const int tl = *time_lag;
    const int s0 = *spinLen, s1 = *trainLen;

    asm volatile("s_wait_asynccnt 0" ::: "memory");
    __syncthreads();                            // ybuf visible to all waves

    // ---- std(y_obs[s0:s1], ddof=1) from LDS ----
    float s = 0.f, q = 0.f;
    for (int j = s0 + tid; j < s1; j += NTHREADS) {
        float v = ybuf[j];
        s += v; q += v * v;
    }
    for (int m = 16; m >= 1; m >>= 1) {
        s += __shfl_xor(s, m, 32);
        q += __shfl_xor(q, m, 32);
    }
    if (lane == 0) { wsum[wave] = s; wsq[wave] = q; }
    __syncthreads();
    if (tid < 32) {   // wave 0 fully active: exact layout-independent WMMA reduce
        float S = wsum[lane], Q = wsq[lane];
        v2f a;  a.x = 1.f; a.y = 1.f;           // A = ones(16x4) -> D rows = colsums(B)
        v2f bs; bs.x = S;  bs.y = S;
        v2f bq; bq.x = Q;  bq.y = Q;
        v8f c = {};
        v8f dS = __builtin_amdgcn_wmma_f32_16x16x4_f32(
            false, a, false, bs, (short)0, c, false, false);
        v8f dQ = __builtin_amdgcn_wmma_f32_16x16x4_f32(
            false, a, false, bq, (short)0, c, false, false);
        float rs = dS[0], rq = dQ[0];           // 32-lane sum of D[0] = 4 * total
        for (int m = 16; m >= 1; m >>= 1) {
            rs += __shfl_xor(rs, m, 32);
            rq += __shfl_xor(rq, m, 32);
        }
        if (lane == 0) {
            float n = (float)(s1 - s0);
            float Ssum = rs * 0.25f, Qsum = rq * 0.25f;
            float var = (Qsum - Ssum * Ssum / n) / (n - 1.f);
            sc[2] = sqrtf(fmaxf(var, 0.f));
            sc[0] = 1.f;                        // carry = identity affine map
            sc[1] = 0.f;
        }
    }
    __syncthreads();

    // ---- streaming chunked scan over b ----
    for (int chunk = 0; chunk < NCHUNK; ++chunk) {
        const int b = chunk * NTHREADS + tid;

        // prefetch next chunk's x rows while we compute (global_prefetch_b8)
        if (chunk + 1 < NCHUNK && (b + NTHREADS) < BB)
            __builtin_prefetch(x + (size_t)(b + NTHREADS) * 20, 0, 3);

        // per-b 10-step affine composition (registers only)
        float F = 1.f, G = 0.f, F9 = 1.f, G9 = 0.f, olL = 0.f, fL = 0.f;
        if (b < BB) {
            const float4* xr = reinterpret_cast<const float4*>(x + (size_t)b * 20);
            float rowv[20];
#pragma unroll
            for (int k = 0; k < 5; ++k) {
                float4 r = xr[k];
                rowv[4 * k + 0] = r.x; rowv[4 * k + 1] = r.y;
                rowv[4 * k + 2] = r.z; rowv[4 * k + 3] = r.w;
            }
            const bool act = (b >= tl);
#pragma unroll
            for (int t = 0; t < TT; ++t) {
                float g  = rowv[2 * t];
                float u2 = rowv[2 * t + 1];
                float ol3 = bias + ((u2 - ML) / SL) * wb2;
                float sig = 1.f / (1.f + expf(-ol3));
                float ol  = ol1 * sig;
                float f   = 1.f - oo - ol;
                if (t == TT - 1) { F9 = F; G9 = G; olL = ol; fL = f; }
                float fe = act ? f : 1.f;
                float ge = act ? g : 0.f;
                G = G * fe + ge;
                F = F * fe;
            }
        }

        // wave-level inclusive affine scan (no barriers)
#pragma unroll
        for (int d = 1; d < 32; d <<= 1) {
            float Fp = __shfl_up(F, d, 32);
            float Gp = __shfl_up(G, d, 32);
            if (lane >= d) { G = Gp * F + G; F = Fp * F; }
        }
        if (lane == 31) { wAF[wave] = F; wAG[wave] = G; }
        // lane-exclusive value
        float eF = __shfl_up(F, 1, 32);
        float eG = __shfl_up(G, 1, 32);
        if (lane == 0) { eF = 1.f; eG = 0.f; }
        __syncthreads();

        // wave 0: scan the 32 wave aggregates, compose global carry
        if (tid < 32) {
            float aF = wAF[lane], aG = wAG[lane];
#pragma unroll
            for (int d = 1; d < 32; d <<= 1) {
                float Fp = __shfl_up(aF, d, 32);
                float Gp = __shfl_up(aG, d, 32);
                if (lane >= d) { aG = Gp * aF + aG; aF = Fp * aF; }
            }
            float exF = __shfl_up(aF, 1, 32);
            float exG = __shfl_up(aG, 1, 32);
            if (lane == 0) { exF = 1.f; exG = 0.f; }
            float cF = sc[0], cG = sc[1];       // reads precede lane31 write (in-order wave)
            wPF[lane] = cF * exF;
            wPG[lane] = cG * exF + exG;
            if (lane == 31) {                   // new carry = carry ∘ chunk total
                sc[0] = cF * aF;
                sc[1] = cG * aF + aG;
            }
        }
        __syncthreads();

        // global exclusive prefix for this b -> incoming state c_start
        float pF = wPF[wave], pG = wPG[wave];
        float gFx = pF * eF;
        float cs  = pG * eF + eG;               // c0 = 0 -> G component only
        (void)gFx;

        if (b < BB) {
            float mask = (b >= tl) ? 1.f : 0.f;
            float c_n = (F9 * cs + G9) * mask;
            float h_n = oo * c_n;
            float l_n = olL * c_n;
            float os  = sc[2] * mask;

            out[0 * BB + b] = h_n;
            out[1 * BB + b] = c_n;
            out[2 * BB + b] = l_n;
            out[3 * BB + b] = 0.f;              // bp_n
            out[4 * BB + b] = 0.f;              // Gate_ib
            out[5 * BB + b] = oo * mask;        // Gate_oo
            out[6 * BB + b] = olL * mask;       // Gate_ol
            out[7 * BB + b] = fL * mask;        // Gate_f
            out[8 * BB + 2 * b + 0] = h_n;      // h_nout[:,0]
            out[8 * BB + 2 * b + 1] = os;       // h_nout[:,1]
            out[10 * BB + b] = os;              // obs_std
        }
    }
}

extern "C" void kernel_launch(void* const* d_in, const int* in_sizes, int n_in,
                              void* d_out, int out_size, void* d_ws, size_t ws_size,
                              hipStream_t stream) {
    const float* x        = (const float*)d_in[0];
    const float* y_obs    = (const float*)d_in[1];
    // d_in[2] = epoch (unused by the math)
    const int*   time_lag = (const int*)d_in[3];
    const float* w_yom    = (const float*)d_in[4];
    const float* w_ylm    = (const float*)d_in[5];
    const float* w_yfm    = (const float*)d_in[6];
    const float* b0_ylm   = (const float*)d_in[7];
    const float* w_b2     = (const float*)d_in[8];
    const int*   spinLen  = (const int*)d_in[9];
    const int*   trainLen = (const int*)d_in[10];

    (void)d_ws; (void)ws_size; (void)in_sizes; (void)n_in; (void)out_size;

    fused_mcpbrnn<<<1, NTHREADS, 0, stream>>>(
        x, y_obs, time_lag, w_yom, w_ylm, w_yfm, b0_ylm, w_b2,
        spinLen, trainLen, (float*)d_out);
}